// VPE_49701361549783
// MI455X (gfx1250) — compile-verified
//
#include <hip/hip_runtime.h>
#include <hip/hip_bf16.h>
#include <math.h>

// Problem constants (from reference): x [32,1536,32,32] f32, proto [32,1536] f32
#define BATCH 32
#define CH    1536
#define HW    1024
#define NP    32
#define TEMP  0.1f
#define EPSN  1e-12f

typedef float v2f __attribute__((ext_vector_type(2)));
typedef float v8f __attribute__((ext_vector_type(8)));

// LDS byte offset of a __shared__-derived pointer: flat LDS addresses keep the
// workgroup-relative byte offset in bits [31:0] (CDNA5 aperture mapping).
__device__ __forceinline__ unsigned lds_off(const void* p) {
    return (unsigned)(uintptr_t)p;
}

// Async copy 16B/lane: global -> LDS, tracked by ASYNCcnt (CDNA5).
__device__ __forceinline__ void async_copy_b128(unsigned lds_byte_off, const void* gptr) {
    asm volatile("global_load_async_to_lds_b128 %0, %1, off"
                 :
                 : "v"(lds_byte_off), "v"((unsigned long long)(uintptr_t)gptr)
                 : "memory");
}

__device__ __forceinline__ void wait_async0() {
    asm volatile("s_wait_asynccnt 0x0" ::: "memory");
}

// ---------------------------------------------------------------------------
// Kernel 1: per (batch, 128-h tile): S = proto[32x1536] * X[1536x128], plus
// column norms, fused softmax over p, write W^T[b][h][p] to workspace.
// Double-buffered async global->LDS staging; WMMA f32 16x16x4, K = c.
// ---------------------------------------------------------------------------
__global__ __launch_bounds__(256)
void vpe_weights_kernel(const float* __restrict__ x,
                        const float* __restrict__ proto,
                        float* __restrict__ wt) {
    const int b     = blockIdx.y;
    const int h_blk = blockIdx.x * 128;

    __shared__ float xs[2][32][136];  // x chunk [c=32][h=128], stride 136 (16B-aligned rows)
    __shared__ float ps[2][32][36];   // proto chunk [p=32][c=32], stride 36

    const int tid  = threadIdx.x;
    const int wave = tid >> 5;
    const int lane = tid & 31;
    const int g    = lane >> 4;   // half-wave group
    const int n    = lane & 15;
    const int hloc = wave * 16 + n;

    // staging mapping (per thread): 1x b128 for proto chunk, 4x b128 for x chunk
    const int sp_p  = tid >> 3;         // proto row 0..31
    const int sp_c4 = (tid & 7) * 4;    // proto col (floats)
    const int sx_col = (tid & 31) * 4;  // x col within 128-h tile
    const int sx_row = tid >> 5;        // x row base (+8j)

    auto stage = [&](int cc, int buf) {
        async_copy_b128(lds_off(&ps[buf][sp_p][sp_c4]),
                        proto + (size_t)sp_p * CH + cc + sp_c4);
        #pragma unroll
        for (int j = 0; j < 4; ++j) {
            const int r = sx_row + j * 8;
            async_copy_b128(lds_off(&xs[buf][r][sx_col]),
                            x + ((size_t)b * CH + cc + r) * HW + h_blk + sx_col);
        }
    };

    v8f acc0 = {}; // S rows p = r + 8g
    v8f acc1 = {}; // S rows p = r + 8g + 16
    float nrm = 0.0f;

    stage(0, 0);  // prologue prefetch of chunk 0

    for (int ci = 0; ci < CH / 32; ++ci) {
        wait_async0();       // this wave's async writes for chunk ci are in LDS
        __syncthreads();     // ...and everyone else's too; prev compute done
        if (ci + 1 < CH / 32) stage((ci + 1) * 32, (ci + 1) & 1);
        const int buf = ci & 1;

        #pragma unroll
        for (int s = 0; s < 8; ++s) {
            const int k = 4 * s + 2 * g;  // this lane's K rows: k, k+1
            const float b0 = xs[buf][k][hloc];
            const float b1 = xs[buf][k + 1][hloc];
            v2f bf; bf[0] = b0; bf[1] = b1;
            v2f a0; a0[0] = ps[buf][n][k];      a0[1] = ps[buf][n][k + 1];
            v2f a1; a1[0] = ps[buf][16 + n][k]; a1[1] = ps[buf][16 + n][k + 1];
            acc0 = __builtin_amdgcn_wmma_f32_16x16x4_f32(false, a0, false, bf,
                                                         (short)0, acc0, false, false);
            acc1 = __builtin_amdgcn_wmma_f32_16x16x4_f32(false, a1, false, bf,
                                                         (short)0, acc1, false, false);
            nrm = fmaf(b0, b0, nrm);   // lane n covers K%4 in {0,1}; lane n+16: {2,3}
            nrm = fmaf(b1, b1, nrm);
        }
    }

    // --- fused softmax over p (32 values per h column) ---
    const float nfull = nrm + __shfl_xor(nrm, 16);
    const float inv   = 1.0f / (fmaxf(sqrtf(nfull), EPSN) * TEMP);

    float m = -INFINITY;
    #pragma unroll
    for (int r = 0; r < 8; ++r) { m = fmaxf(m, acc0[r]); m = fmaxf(m, acc1[r]); }
    m = fmaxf(m, __shfl_xor(m, 16));
    const float ms = m * inv;

    float e0[8], e1[8];
    float sum = 0.0f;
    #pragma unroll
    for (int r = 0; r < 8; ++r) {
        e0[r] = expf(acc0[r] * inv - ms); sum += e0[r];
        e1[r] = expf(acc1[r] * inv - ms); sum += e1[r];
    }
    sum += __shfl_xor(sum, 16);
    const float rs = 1.0f / sum;

    float* wrow = wt + ((size_t)b * HW + h_blk + hloc) * NP;
    #pragma unroll
    for (int r = 0; r < 8; ++r) {
        wrow[r + 8 * g]      = e0[r] * rs;
        wrow[r + 8 * g + 16] = e1[r] * rs;
    }
}

// ---------------------------------------------------------------------------
// Kernel 2: per (batch, 128-c tile): agg[32 x 128] = W[32x1024] * X^T[1024x128].
// Double-buffered async global->LDS staging; WMMA f32 16x16x4, K = h.
// ---------------------------------------------------------------------------
__global__ __launch_bounds__(256)
void vpe_aggregate_kernel(const float* __restrict__ x,
                          const float* __restrict__ wt,
                          float* __restrict__ out) {
    const int b    = blockIdx.y;
    const int cblk = blockIdx.x * 128;

    __shared__ float xs2[2][128][36]; // x chunk [c=128][h=32], stride 36
    __shared__ float ws2[2][32][40];  // W^T chunk [h=32][p=32], stride 40

    const int tid  = threadIdx.x;
    const int wave = tid >> 5;
    const int lane = tid & 31;
    const int g    = lane >> 4;
    const int n    = lane & 15;
    const int cloc = wave * 16 + n;

    const int sw_h  = tid >> 3;        // W^T row 0..31
    const int sw_p4 = (tid & 7) * 4;   // W^T col (floats)
    const int sx_row = tid >> 3;       // x row base (+32j)
    const int sx_col = (tid & 7) * 4;  // x col within 32-h chunk

    auto stage = [&](int hh, int buf) {
        async_copy_b128(lds_off(&ws2[buf][sw_h][sw_p4]),
                        wt + ((size_t)b * HW + hh + sw_h) * NP + sw_p4);
        #pragma unroll
        for (int j = 0; j < 4; ++j) {
            const int row = sx_row + 32 * j;
            async_copy_b128(lds_off(&xs2[buf][row][sx_col]),
                            x + ((size_t)b * CH + cblk + row) * HW + hh + sx_col);
        }
    };

    v8f acc0 = {}; // rows p = r + 8g
    v8f acc1 = {}; // rows p = r + 8g + 16

    stage(0, 0);

    for (int ci = 0; ci < HW / 32; ++ci) {
        wait_async0();
        __syncthreads();
        if (ci + 1 < HW / 32) stage((ci + 1) * 32, (ci + 1) & 1);
        const int buf = ci & 1;

        #pragma unroll
        for (int s = 0; s < 8; ++s) {
            const int k = 4 * s + 2 * g;
            v2f bf; bf[0] = xs2[buf][cloc][k]; bf[1] = xs2[buf][cloc][k + 1];
            v2f a0; a0[0] = ws2[buf][k][n];      a0[1] = ws2[buf][k + 1][n];
            v2f a1; a1[0] = ws2[buf][k][16 + n]; a1[1] = ws2[buf][k + 1][16 + n];
            acc0 = __builtin_amdgcn_wmma_f32_16x16x4_f32(false, a0, false, bf,
                                                         (short)0, acc0, false, false);
            acc1 = __builtin_amdgcn_wmma_f32_16x16x4_f32(false, a1, false, bf,
                                                         (short)0, acc1, false, false);
        }
    }

    // store out[b][p][cblk+cloc]; each half-wave writes 64B-contiguous runs
    float* obase = out + (size_t)b * NP * CH + cblk + cloc;
    #pragma unroll
    for (int r = 0; r < 8; ++r) {
        obase[(size_t)(r + 8 * g) * CH]      = acc0[r];
        obase[(size_t)(r + 8 * g + 16) * CH] = acc1[r];
    }
}

extern "C" void kernel_launch(void* const* d_in, const int* in_sizes, int n_in,
                              void* d_out, int out_size, void* d_ws, size_t ws_size,
                              hipStream_t stream) {
    (void)in_sizes; (void)n_in; (void)out_size; (void)ws_size;
    const float* x     = (const float*)d_in[0];   // [32,1536,32,32] f32
    const float* proto = (const float*)d_in[1];   // [32,1536,1,1]  f32
    float* out = (float*)d_out;                   // [32,32,1536]   f32
    float* wt  = (float*)d_ws;                    // W^T[b][hw][p]: 4 MB scratch

    dim3 grid1(HW / 128, BATCH);   // 8 x 32 blocks
    vpe_weights_kernel<<<grid1, 256, 0, stream>>>(x, proto, wt);

    dim3 grid2(CH / 128, BATCH);   // 12 x 32 blocks
    vpe_aggregate_kernel<<<grid2, 256, 0, stream>>>(x, wt, out);
}